// AttentionGNN_9715216024204
// MI455X (gfx1250) — compile-verified
//
#include <hip/hip_runtime.h>
#include <hip/hip_bf16.h>
#include <math.h>

// ---------------------------------------------------------------------------
// AttentionGNN (3x GATConv, 4 heads x 32ch, HID=128) for MI455X / gfx1250.
// GEMMs run on V_WMMA_F32_16X16X4_F32 (fp32 A/B, fp32 acc) — full precision,
// wave32 fragment layouts per CDNA5 ISA 7.12.2. Edge softmax/aggregation is
// L2-resident float atomics (all node arrays ~51MB each fit the 192MB L2).
// ---------------------------------------------------------------------------

typedef __attribute__((ext_vector_type(2))) float v2f;
typedef __attribute__((ext_vector_type(8))) float v8f;

#define HID 128
#define HEADS 4
#define CH 32
#define NEG_SLOPE 0.2f

// ---------- fp32 WMMA GEMM: out[N,128] = in[N,128] @ W[128,128] + bias ------
// Block = 256 threads = 8 waves; each wave computes one 16-row tile x 128 cols.
// W staged in LDS (64KB). A fragments preloaded into 32 v2f registers.
__global__ __launch_bounds__(256) void gemm128_wmma(
    const float* __restrict__ in, const float* __restrict__ W,
    const float* __restrict__ bias, float* __restrict__ out, int ntiles) {
  __shared__ float Wlds[HID * HID];
  const int tid = threadIdx.x;
  for (int i = tid; i < HID * HID; i += 256) Wlds[i] = W[i];
  __syncthreads();

  const int wave = tid >> 5;
  const int lane = tid & 31;
  const int tile = blockIdx.x * 8 + wave;
  if (tile >= ntiles) return;           // wave-uniform: EXEC stays all-ones

  const int half = lane >> 4;           // 0: K=k,k+1   1: K=k+2,k+3
  const int lrow = lane & 15;           // A row / B,D column within tile
  const long row = (long)tile * 16 + lrow;
  const float* arow = in + row * HID;

  // A: 16x4 fp32 fragment per K-step (ISA 7.12.2, 32-bit A-matrix 16x4)
  v2f afrag[32];
#pragma unroll
  for (int k = 0; k < 32; ++k) {
    const int kk = k * 4 + half * 2;
    afrag[k].x = arow[kk];
    afrag[k].y = arow[kk + 1];
  }

  for (int nt = 0; nt < 8; ++nt) {
    const int col = nt * 16 + lrow;
    v8f acc = {0.f, 0.f, 0.f, 0.f, 0.f, 0.f, 0.f, 0.f};
#pragma unroll
    for (int k = 0; k < 32; ++k) {
      const int kk = k * 4 + half * 2;
      v2f b;
      b.x = Wlds[kk * HID + col];
      b.y = Wlds[(kk + 1) * HID + col];
      acc = __builtin_amdgcn_wmma_f32_16x16x4_f32(
          false, afrag[k], false, b, (short)0, acc, false, false);
    }
    const float bcol = bias[col];
#pragma unroll
    for (int r = 0; r < 8; ++r) {
      const long orow = (long)tile * 16 + r + half * 8;   // D: VGPR r -> M=r / r+8
      out[orow * HID + col] = acc[r] + bcol;
    }
  }
}

// ---------- attention logit terms per node/head -----------------------------
__global__ void compute_al(const float* __restrict__ hW,
                           const float* __restrict__ asrc,
                           const float* __restrict__ adst,
                           float* __restrict__ alS, float* __restrict__ alD,
                           int n) {
  const int idx = blockIdx.x * blockDim.x + threadIdx.x;
  if (idx >= n * HEADS) return;
  const int node = idx >> 2, hh = idx & 3;
  const float* hrow = hW + (long)node * HID + hh * CH;
  float ss = 0.f, dd = 0.f;
#pragma unroll 8
  for (int c = 0; c < CH; ++c) {
    const float v = hrow[c];
    ss += v * asrc[hh * CH + c];
    dd += v * adst[hh * CH + c];
  }
  alS[idx] = ss;
  alD[idx] = dd;
}

// ---------- per-layer state init -------------------------------------------
__global__ void init_layer(float* __restrict__ m, float* __restrict__ s,
                           float* __restrict__ acc, int n) {
  const int idx = blockIdx.x * blockDim.x + threadIdx.x;
  if (idx < n * HID) acc[idx] = 0.f;
  if (idx < n * HEADS) { m[idx] = -3.0e38f; s[idx] = 0.f; }
}

__device__ __forceinline__ float leaky(float x) {
  return x > 0.f ? x : NEG_SLOPE * x;
}

__device__ __forceinline__ void atomicMaxFloat(float* addr, float val) {
  // IEEE ordering trick: positive floats order as ints, negative as reversed uints.
  if (val >= 0.f)
    atomicMax((int*)addr, __float_as_int(val));
  else
    atomicMin((unsigned int*)addr, __float_as_uint(val));
}

// ---------- pass A: segment max of leaky-relu logits over dst ---------------
__global__ void edge_max(const int* __restrict__ ei,
                         const float* __restrict__ alS,
                         const float* __restrict__ alD,
                         float* __restrict__ m, int E, int n) {
  const int idx = blockIdx.x * blockDim.x + threadIdx.x;
  const int etot = E + n;
  if (idx >= etot * HEADS) return;
  const int e = idx >> 2, hh = idx & 3;
  int src, dst;
  if (e < E) { src = ei[e]; dst = ei[E + e]; } else { src = dst = e - E; }
  const float logit = leaky(alS[src * HEADS + hh] + alD[dst * HEADS + hh]);
  atomicMaxFloat(&m[dst * HEADS + hh], logit);
}

// ---------- pass B: fused exp-sum + weighted feature scatter ----------------
// One block (128 threads) per edge; thread t = channel, head = t>>5.
__global__ __launch_bounds__(128) void edge_accum(
    const int* __restrict__ ei, const float* __restrict__ alS,
    const float* __restrict__ alD, const float* __restrict__ m,
    const float* __restrict__ hW, float* __restrict__ s,
    float* __restrict__ acc, int E, int n) {
  const int e = blockIdx.x;
  const int c = threadIdx.x;
  const int hh = c >> 5;
  int src, dst;
  if (e < E) { src = ei[e]; dst = ei[E + e]; } else { src = dst = e - E; }
  const float logit = leaky(alS[src * HEADS + hh] + alD[dst * HEADS + hh]);
  const float ev = __expf(logit - m[dst * HEADS + hh]);
  if ((c & 31) == 0) atomicAdd(&s[dst * HEADS + hh], ev);
  atomicAdd(&acc[(long)dst * HID + c], hW[(long)src * HID + c] * ev);
}

// ---------- finalize (concat layers): normalize + bias + ELU ----------------
__global__ void finalize_concat(const float* __restrict__ acc,
                                const float* __restrict__ s,
                                const float* __restrict__ bias,
                                float* __restrict__ out, int n) {
  const int idx = blockIdx.x * blockDim.x + threadIdx.x;
  if (idx >= n * HID) return;
  const int node = idx >> 7, c = idx & 127, hh = c >> 5;
  float v = acc[idx] / (s[node * HEADS + hh] + 1e-16f) + bias[c];
  out[idx] = v > 0.f ? v : expm1f(v);
}

// ---------- finalize (mean layer): head-mean + bias + ELU -> [N,32] ---------
__global__ void finalize_mean(const float* __restrict__ acc,
                              const float* __restrict__ s,
                              const float* __restrict__ b2,
                              float* __restrict__ out32, int n) {
  const int idx = blockIdx.x * blockDim.x + threadIdx.x;
  if (idx >= n * CH) return;
  const int node = idx >> 5, c = idx & 31;
  float sum = 0.f;
#pragma unroll
  for (int hh = 0; hh < HEADS; ++hh)
    sum += acc[(long)node * HID + hh * CH + c] / (s[node * HEADS + hh] + 1e-16f);
  float v = sum * 0.25f + b2[c];
  out32[idx] = v > 0.f ? v : expm1f(v);
}

// ---------- graph pooling ---------------------------------------------------
__global__ void pool_zero(float* __restrict__ pooled, float* __restrict__ cnt,
                          int b) {
  const int idx = blockIdx.x * blockDim.x + threadIdx.x;
  if (idx < b * CH) pooled[idx] = 0.f;
  if (idx < b) cnt[idx] = 0.f;
}

__global__ void pool_accum(const float* __restrict__ h32,
                           const int* __restrict__ batch,
                           float* __restrict__ pooled, float* __restrict__ cnt,
                           int n) {
  const int node = blockIdx.x * blockDim.x + threadIdx.x;
  if (node >= n) return;
  const int b = batch[node];
  atomicAdd(&cnt[b], 1.f);
#pragma unroll 8
  for (int c = 0; c < CH; ++c)
    atomicAdd(&pooled[b * CH + c], h32[(long)node * CH + c]);
}

// ---------- MLP head: one block (128 threads) per graph ---------------------
__global__ __launch_bounds__(128) void mlp_head(
    const float* __restrict__ pooled, const float* __restrict__ cnt,
    const float* __restrict__ w_fp, const float* __restrict__ b_fp,
    const float* __restrict__ w_m1, const float* __restrict__ b_m1,
    const float* __restrict__ w_m2, const float* __restrict__ b_m2,
    float* __restrict__ out) {
  __shared__ float p[CH];
  __shared__ float o1[HID];
  __shared__ float hm[HID / 2];
  const int b = blockIdx.x;
  const int t = threadIdx.x;
  if (t < CH) p[t] = pooled[b * CH + t] / fmaxf(cnt[b], 1.f);
  __syncthreads();
  {
    float a = b_fp[t];
#pragma unroll 8
    for (int c = 0; c < CH; ++c) a += p[c] * w_fp[c * HID + t];
    o1[t] = a;
  }
  __syncthreads();
  if (t < HID / 2) {
    float a = b_m1[t];
#pragma unroll 8
    for (int j = 0; j < HID; ++j) a += o1[j] * w_m1[j * (HID / 2) + t];
    hm[t] = fmaxf(a, 0.f);
  }
  __syncthreads();
  if (t == 0) {
    float a = b_m2[0];
    for (int k = 0; k < HID / 2; ++k) a += hm[k] * w_m2[k];
    out[b] = a;
  }
}

// ---------------------------------------------------------------------------
extern "C" void kernel_launch(void* const* d_in, const int* in_sizes, int n_in,
                              void* d_out, int out_size, void* d_ws,
                              size_t ws_size, hipStream_t stream) {
  const float* x     = (const float*)d_in[0];
  const int*   ei    = (const int*)d_in[1];
  const int*   batch = (const int*)d_in[2];
  const float* w_in  = (const float*)d_in[3];
  const float* b_in  = (const float*)d_in[4];
  const float* wL[3]    = {(const float*)d_in[5],  (const float*)d_in[9],  (const float*)d_in[13]};
  const float* asrcL[3] = {(const float*)d_in[6],  (const float*)d_in[10], (const float*)d_in[14]};
  const float* adstL[3] = {(const float*)d_in[7],  (const float*)d_in[11], (const float*)d_in[15]};
  const float* bL[3]    = {(const float*)d_in[8],  (const float*)d_in[12], (const float*)d_in[16]};
  const float* w_fp = (const float*)d_in[17];
  const float* b_fp = (const float*)d_in[18];
  const float* w_m1 = (const float*)d_in[19];
  const float* b_m1 = (const float*)d_in[20];
  const float* w_m2 = (const float*)d_in[21];
  const float* b_m2 = (const float*)d_in[22];
  float* outp = (float*)d_out;

  const int N = in_sizes[0] / HID;   // 100000 (multiple of 16)
  const int E = in_sizes[1] / 2;     // 640000
  const int B = out_size;            // 256
  const int Etot = E + N;            // self-loops appended

  // workspace layout (floats)
  float* ws = (float*)d_ws;
  size_t o = 0;
  float* buf0 = ws + o; o += (size_t)N * HID;   // current features
  float* buf1 = ws + o; o += (size_t)N * HID;   // h @ W
  float* buf2 = ws + o; o += (size_t)N * HID;   // edge-weighted accumulator
  float* alS  = ws + o; o += (size_t)N * HEADS;
  float* alD  = ws + o; o += (size_t)N * HEADS;
  float* segm = ws + o; o += (size_t)N * HEADS;
  float* segs = ws + o; o += (size_t)N * HEADS;
  float* h32  = ws + o; o += (size_t)N * CH;
  float* pooled = ws + o; o += (size_t)B * CH;
  float* cnt    = ws + o; o += (size_t)B;
  (void)ws_size; (void)n_in;

  const int ntiles = N / 16;
  const int gemm_blocks = (ntiles + 7) / 8;

  // input linear: buf0 = x @ w_in + b_in
  gemm128_wmma<<<gemm_blocks, 256, 0, stream>>>(x, w_in, b_in, buf0, ntiles);

  for (int layer = 0; layer < 3; ++layer) {
    // buf1 = buf0 @ W_layer   (bias of GAT linear is implicit zero in reference;
    // layer bias bL is applied post-aggregation, so pass a zero: reuse segs
    // pre-init trick is unsafe -> use a dedicated zeroed bias = first 128 of buf2
    // after init_layer (acc starts zeroed). Order: init first, then gemm.)
    init_layer<<<((size_t)N * HID + 255) / 256, 256, 0, stream>>>(segm, segs, buf2, N);
    gemm128_wmma<<<gemm_blocks, 256, 0, stream>>>(buf0, wL[layer], buf2 /*zero bias*/, buf1, ntiles);
    compute_al<<<(N * HEADS + 255) / 256, 256, 0, stream>>>(
        buf1, asrcL[layer], adstL[layer], alS, alD, N);
    edge_max<<<((size_t)Etot * HEADS + 255) / 256, 256, 0, stream>>>(
        ei, alS, alD, segm, E, N);
    edge_accum<<<Etot, 128, 0, stream>>>(ei, alS, alD, segm, buf1, segs, buf2, E, N);
    if (layer < 2) {
      finalize_concat<<<((size_t)N * HID + 255) / 256, 256, 0, stream>>>(
          buf2, segs, bL[layer], buf0, N);
    } else {
      finalize_mean<<<((size_t)N * CH + 255) / 256, 256, 0, stream>>>(
          buf2, segs, bL[layer], h32, N);
    }
  }

  // mean pooling over batch segments
  pool_zero<<<(B * CH + 255) / 256, 256, 0, stream>>>(pooled, cnt, B);
  pool_accum<<<(N + 255) / 256, 256, 0, stream>>>(h32, batch, pooled, cnt, N);

  // final projection + MLP head
  mlp_head<<<B, 128, 0, stream>>>(pooled, cnt, w_fp, b_fp, w_m1, b_m1, w_m2,
                                  b_m2, outp);
}